// GraphWaveletConv_20151986553557
// MI455X (gfx1250) — compile-verified
//
#include <hip/hip_runtime.h>
#include <hip/hip_bf16.h>

// ---------------- problem constants ----------------
#define N_NODES 50000
#define N_EDGES 800000
#define D_IN    64
#define D_OUT_TOT 256   // SCALES * D_OUT = 4*64

// ---------------- scratch layout (in floats) ----------------
// [0..1]                : minbits, maxbits (as uint)
// [64 .. 64+50048)      : deg[N]
// [50112 .. 50112+50048): dis[N]  (deg^-1/2)
// [100352 .. +12.8M)    : H[N, 256]  fp32 GEMM result (51.2 MB, L2-resident)
#define WS_MM   0
#define WS_DEG  64
#define WS_DIS  50112
#define WS_H    100352

typedef __attribute__((ext_vector_type(2))) float v2f;
typedef __attribute__((ext_vector_type(8))) float v8f;

// ---------------- 1) init: minmax seeds + deg = 1 (self loop weight) ----
__global__ void init_kernel(float* ws) {
    int i = blockIdx.x * blockDim.x + threadIdx.x;
    if (i == 0) {
        unsigned* mm = (unsigned*)(ws + WS_MM);
        mm[0] = 0x7F800000u;  // +inf bits (min accumulator)
        mm[1] = 0u;           // 0.0f bits (max accumulator; weights >= 0)
    }
    if (i < N_NODES) ws[WS_DEG + i] = 1.0f;  // self-loop contributes 1
}

// ---------------- 2) min/max reduction over edge weights ----------------
__global__ void minmax_kernel(const float* __restrict__ ew, float* ws) {
    __shared__ float smin[256];
    __shared__ float smax[256];
    int tid = threadIdx.x;
    float lmin = __builtin_inff();
    float lmax = 0.0f;  // weights are uniform [0,1): nonnegative
    for (int e = blockIdx.x * blockDim.x + tid; e < N_EDGES;
         e += gridDim.x * blockDim.x) {
        float w = ew[e];
        lmin = fminf(lmin, w);
        lmax = fmaxf(lmax, w);
    }
    smin[tid] = lmin; smax[tid] = lmax;
    __syncthreads();
    for (int s = 128; s > 0; s >>= 1) {
        if (tid < s) {
            smin[tid] = fminf(smin[tid], smin[tid + s]);
            smax[tid] = fmaxf(smax[tid], smax[tid + s]);
        }
        __syncthreads();
    }
    if (tid == 0) {
        unsigned* mm = (unsigned*)(ws + WS_MM);
        // bitwise compare == float compare for nonnegative floats
        atomicMin(&mm[0], __float_as_uint(smin[0]));
        atomicMax(&mm[1], __float_as_uint(smax[0]));
    }
}

// ---------------- 3) degree scatter-add of normalized weights ----------
__global__ void degree_kernel(const long long* __restrict__ ei,
                              const float* __restrict__ ew, float* ws) {
    int e = blockIdx.x * blockDim.x + threadIdx.x;
    if (e >= N_EDGES) return;
    const unsigned* mm = (const unsigned*)(ws + WS_MM);
    float wmin = __uint_as_float(mm[0]);
    float wmax = __uint_as_float(mm[1]);
    float w = (ew[e] - wmin) / (wmax - wmin);
    int c = (int)ei[N_EDGES + e];  // edge_index[1][e]
    atomicAdd(ws + WS_DEG + c, w);
}

// ---------------- 4) dis = deg^-1/2 ------------------------------------
__global__ void dis_kernel(float* ws) {
    int i = blockIdx.x * blockDim.x + threadIdx.x;
    if (i >= N_NODES) return;
    float d = ws[WS_DEG + i];
    ws[WS_DIS + i] = (d > 0.0f) ? rsqrtf(fmaxf(d, 1e-12f)) : 0.0f;
}

// ---------------- 5) WMMA FP32 GEMM: H[N,256] = x[N,64] @ B[64,256] ----
// B[k, s*64+n] = Ws[s][k][n]  (Ws flat: s*4096 + k*64 + n)
// One wave per 16x16 output tile. 16 chained V_WMMA_F32_16X16X4_F32 (K=4 each).
__global__ __launch_bounds__(256) void gemm_kernel(
    const float* __restrict__ x, const float* __restrict__ Ws,
    float* __restrict__ ws) {
    float* H = ws + WS_H;
    int lane  = threadIdx.x & 31;
    int waveI = threadIdx.x >> 5;
    int tile  = blockIdx.x * 8 + waveI;       // 6250 blocks * 8 waves = 50000 tiles
    int nTile = tile & 15;                    // 16 tiles across 256 cols
    int mTile = tile >> 4;                    // 3125 tiles down 50000 rows
    int m0 = mTile * 16;
    int n0 = nTile * 16;

    int half = lane >> 4;        // 0: K even-pair base, 1: K+2
    int l16  = lane & 15;

    // A: row m0+l16 of x; per k-step this lane holds x[row][kk], x[row][kk+1]
    const float* xrow = x + (size_t)(m0 + l16) * D_IN;
    // B: column n0+l16 of packed W
    int col = n0 + l16;
    const float* Bcol = Ws + (size_t)(col >> 6) * 4096 + (col & 63);

    v8f acc = {};
    #pragma unroll
    for (int k = 0; k < D_IN; k += 4) {
        int kk = k + 2 * half;
        v2f a;
        a.x = xrow[kk];
        a.y = xrow[kk + 1];
        v2f b;
        b.x = Bcol[kk * D_OUT_TOT / 4 * 0 + (size_t)kk * 64];       // B[kk, col]
        b.y = Bcol[(size_t)(kk + 1) * 64];                          // B[kk+1, col]
        // 8 args: (neg_a, A, neg_b, B, c_mod, C, reuse_a, reuse_b)
        acc = __builtin_amdgcn_wmma_f32_16x16x4_f32(
            false, a, false, b, (short)0, acc, false, false);
    }

    // D layout: VGPR v -> row m0 + v + 8*half, col n0 + l16
    float* Hout = H + (size_t)col;
    #pragma unroll
    for (int v = 0; v < 8; ++v) {
        Hout[(size_t)(m0 + v + 8 * half) * D_OUT_TOT] = acc[v];
    }
}

// ---------------- 6) out = dis^2 * H + bias (self loop) -----------------
__global__ void outinit_kernel(const float* __restrict__ bs,
                               const float* __restrict__ ws,
                               float* __restrict__ out) {
    int idx = blockIdx.x * blockDim.x + threadIdx.x;  // N*256 threads
    if (idx >= N_NODES * D_OUT_TOT) return;
    int n = idx >> 8;
    int c = idx & 255;
    float dis = ws[WS_DIS + n];
    out[idx] = dis * dis * ws[WS_H + idx] + bs[c];
}

// ---------------- 7) edge scatter: one wave per edge --------------------
__global__ __launch_bounds__(256) void scatter_kernel(
    const long long* __restrict__ ei, const float* __restrict__ ew,
    const float* __restrict__ ws, float* __restrict__ out) {
    int gid  = blockIdx.x * blockDim.x + threadIdx.x;
    int e    = gid >> 5;          // edge id (one wave32 per edge)
    int lane = threadIdx.x & 31;  // lane covers 8 of 256 features
    if (e >= N_EDGES) return;

    int r = (int)ei[e];
    int c = (int)ei[N_EDGES + e];
    const unsigned* mm = (const unsigned*)(ws + WS_MM);
    float wmin = __uint_as_float(mm[0]);
    float wmax = __uint_as_float(mm[1]);
    float w = (ew[e] - wmin) / (wmax - wmin);
    float norm = ws[WS_DIS + r] * w * ws[WS_DIS + c];

    const float4* hp = (const float4*)(ws + WS_H + (size_t)r * D_OUT_TOT);
    float4 h0 = hp[lane * 2];
    float4 h1 = hp[lane * 2 + 1];
    float* op = out + (size_t)c * D_OUT_TOT + lane * 8;
    atomicAdd(op + 0, norm * h0.x);
    atomicAdd(op + 1, norm * h0.y);
    atomicAdd(op + 2, norm * h0.z);
    atomicAdd(op + 3, norm * h0.w);
    atomicAdd(op + 4, norm * h1.x);
    atomicAdd(op + 5, norm * h1.y);
    atomicAdd(op + 6, norm * h1.z);
    atomicAdd(op + 7, norm * h1.w);
}

extern "C" void kernel_launch(void* const* d_in, const int* in_sizes, int n_in,
                              void* d_out, int out_size, void* d_ws, size_t ws_size,
                              hipStream_t stream) {
    const float*     x  = (const float*)d_in[0];
    const long long* ei = (const long long*)d_in[1];  // int64 edge_index [2,E]
    const float*     ew = (const float*)d_in[2];
    const float*     Ws = (const float*)d_in[3];      // [4,64,64]
    const float*     bs = (const float*)d_in[4];      // [4,64] flat = [256]
    float* out = (float*)d_out;
    float* ws  = (float*)d_ws;
    (void)in_sizes; (void)n_in; (void)out_size; (void)ws_size;

    // 1) init minmax + deg
    init_kernel<<<(N_NODES + 255) / 256, 256, 0, stream>>>(ws);
    // 2) min/max of edge weights
    minmax_kernel<<<512, 256, 0, stream>>>(ew, ws);
    // 3) degree
    degree_kernel<<<(N_EDGES + 255) / 256, 256, 0, stream>>>(ei, ew, ws);
    // 4) deg^-1/2
    dis_kernel<<<(N_NODES + 255) / 256, 256, 0, stream>>>(ws);
    // 5) H = x @ [W0|W1|W2|W3]  (WMMA fp32, 8 waves/block, 1 tile/wave)
    gemm_kernel<<<(N_NODES / 16) * (D_OUT_TOT / 16) / 8, 256, 0, stream>>>(x, Ws, ws);
    // 6) out = dis^2 * H + bias
    outinit_kernel<<<(N_NODES * D_OUT_TOT + 255) / 256, 256, 0, stream>>>(bs, ws, out);
    // 7) edge scatter-add
    scatter_kernel<<<(N_EDGES * 32) / 256, 256, 0, stream>>>(ei, ew, ws, out);
}